// LossCorrsSparse_25881472925979
// MI455X (gfx1250) — compile-verified
//
#include <hip/hip_runtime.h>
#include <hip/hip_bf16.h>
#include <math.h>

// Problem constants (match reference)
#define NB_  32
#define RES_ 256
#define KP_  1024
#define NK_  (NB_ * KP_)
#define EPS_ 1e-6f
#define INV_T_ 1.0f

typedef float v2f __attribute__((ext_vector_type(2)));
typedef float v8f __attribute__((ext_vector_type(8)));

// Fast hardware sqrt: input is clamped to [0, +inf) and finite, so the single
// v_sqrt_f32 (~1 ulp) is adequate; avoids the IEEE correction expansion.
__device__ inline float fast_sqrtf(float x) { return __builtin_amdgcn_sqrtf(x); }

// ---------------------------------------------------------------------------
// Phase A: bilinear sampling of flow at keypoints. Emits WMMA-ready payloads:
//   A2   [NK,2] = -2*(src+EPS)            (A-matrix K0/K1, lanes 0-15)
//   PADNA[NK,2] = (||src+EPS||^2, 1.0)    (A-matrix K2/K3, lanes 16-31)
//   Bm   [NK,2] = trg                     (B-matrix K0/K1, lanes 0-15)
//   PADNB[NK,2] = (1.0, ||trg||^2)        (B-matrix K2/K3, lanes 16-31)
// With A scaled by -2 and norms in the spare K slots, one WMMA with C=0
// produces the full 16x16 tile of SQUARED DISTANCES directly.
// ---------------------------------------------------------------------------
__device__ inline void gather2(const float* __restrict__ f, float xi, float yi,
                               float& gx, float& gy) {
    float valid = (xi >= 0.f && xi <= (float)(RES_ - 1) &&
                   yi >= 0.f && yi <= (float)(RES_ - 1)) ? 1.f : 0.f;
    int xc = (int)fminf(fmaxf(xi, 0.f), (float)(RES_ - 1));
    int yc = (int)fminf(fmaxf(yi, 0.f), (float)(RES_ - 1));
    const float* p = f + ((size_t)yc * RES_ + (size_t)xc) * 2;
    gx = p[0] * valid;
    gy = p[1] * valid;
}

__device__ inline void bilin2(const float* __restrict__ f, float x, float y,
                              float& ox, float& oy) {
    float x0 = floorf(x), y0 = floorf(y);
    float wx = x - x0, wy = y - y0;
    float ax, ay, bx, by, cx, cy, dx, dy;
    gather2(f, x0,       y0,       ax, ay);
    gather2(f, x0 + 1.f, y0,       bx, by);
    gather2(f, x0,       y0 + 1.f, cx, cy);
    gather2(f, x0 + 1.f, y0 + 1.f, dx, dy);
    float w00 = (1.f - wx) * (1.f - wy);
    float w10 = wx * (1.f - wy);
    float w01 = (1.f - wx) * wy;
    float w11 = wx * wy;
    ox = ax * w00 + bx * w10 + cx * w01 + dx * w11;
    oy = ay * w00 + by * w10 + cy * w01 + dy * w11;
}

__global__ __launch_bounds__(256) void sample_kernel(
    const float* __restrict__ srcF, const float* __restrict__ trgF,
    const float* __restrict__ srcK, const float* __restrict__ trgK,
    float* __restrict__ A2, float* __restrict__ Bm,
    float* __restrict__ PADNA, float* __restrict__ PADNB) {
    int t = blockIdx.x * blockDim.x + threadIdx.x;
    if (t >= NK_) return;
    int n = t / KP_;
    const float* sfl = srcF + (size_t)n * RES_ * RES_ * 2;
    const float* tfl = trgF + (size_t)n * RES_ * RES_ * 2;

    float sx, sy, tx, ty;
    bilin2(sfl, srcK[(size_t)t * 2], srcK[(size_t)t * 2 + 1], sx, sy);
    bilin2(tfl, trgK[(size_t)t * 2], trgK[(size_t)t * 2 + 1], tx, ty);

    sx += EPS_; sy += EPS_;                 // eps added to the difference == added to src side
    A2[(size_t)t * 2]        = -2.f * sx;
    A2[(size_t)t * 2 + 1]    = -2.f * sy;
    PADNA[(size_t)t * 2]     = sx * sx + sy * sy;
    PADNA[(size_t)t * 2 + 1] = 1.0f;
    Bm[(size_t)t * 2]        = tx;
    Bm[(size_t)t * 2 + 1]    = ty;
    PADNB[(size_t)t * 2]     = 1.0f;
    PADNB[(size_t)t * 2 + 1] = tx * tx + ty * ty;
}

// ---------------------------------------------------------------------------
// Phase B: one wave per (batch n, 16-column strip). 64 row-tile iterations,
// each: one uniform global_load_b64 + one v_wmma_f32_16x16x4_f32 giving the
// 16x16 squared-distance tile, then branch-free online logsumexp per column.
// Diagonal distance computed analytically once (not in the loop). Halves
// merged at the end via lane-xor-16 shuffle (wave32).
// ---------------------------------------------------------------------------
__global__ __launch_bounds__(256) void pair_lse_kernel(
    const float* __restrict__ A2, const float* __restrict__ Bm,
    const float* __restrict__ PADNA, const float* __restrict__ PADNB,
    const unsigned char* __restrict__ vis, const float* __restrict__ wt,
    float* __restrict__ LOSS) {
    int wave = (blockIdx.x * blockDim.x + threadIdx.x) >> 5;   // 0..2047
    int lane = threadIdx.x & 31;
    int n     = wave >> 6;          // 64 column tiles per batch
    int jTile = wave & 63;
    int jBase = jTile << 4;
    int half  = lane >> 4;          // 0: K0/K1 payload lanes, 1: K2/K3 payload lanes
    int l     = lane & 15;
    int jcol  = jBase + l;

    const float* An  = A2    + (size_t)n * KP_ * 2;
    const float* Bn  = Bm    + (size_t)n * KP_ * 2;
    const float* PAn = PADNA + (size_t)n * KP_ * 2;
    const float* PBn = PADNB + (size_t)n * KP_ * 2;

    // Uniform (branch-free) per-lane source pointers: low lanes read the -2a /
    // b pairs, high lanes read the (norm, 1) / (1, norm) pairs. Same stride.
    const float* aLane = (half ? PAn : An) + (size_t)l * 2;
    const float* bLane = (half ? PBn : Bn) + (size_t)jcol * 2;

    v2f bReg = *(const v2f*)bLane;   // loop invariant

    // Diagonal distance for this column, computed directly:
    //   d_jj^2 = ||a_j||^2 + ||b_j||^2 + (-2 a_j).b_j
    float dj2 = PAn[(size_t)jcol * 2] + PBn[(size_t)jcol * 2 + 1]
              + An[(size_t)jcol * 2]     * Bn[(size_t)jcol * 2]
              + An[(size_t)jcol * 2 + 1] * Bn[(size_t)jcol * 2 + 1];
    float dj = fast_sqrtf(fmaxf(dj2, 0.f));

    float m = -INFINITY;   // running column max of (-dist/T)
    float s = 0.f;         // running column sum of exp(x - m)

    for (int iBase = 0; iBase < KP_; iBase += 16) {
        v2f aReg = *(const v2f*)(aLane + (size_t)iBase * 2);

        v8f c = {0.f, 0.f, 0.f, 0.f, 0.f, 0.f, 0.f, 0.f};
        // D[r] = squared distance for row M = r + 8*half, column N = jcol
        v8f d2 = __builtin_amdgcn_wmma_f32_16x16x4_f32(
            false, aReg, false, bReg, (short)0, c, false, false);

        float x[8];
        float tm = -INFINITY;
        #pragma unroll
        for (int r = 0; r < 8; ++r) {
            float dd = fmaxf(d2[r], 0.f);
            x[r] = -fast_sqrtf(dd) * INV_T_;   // single v_sqrt_f32 (TRANS)
            tm   = fmaxf(tm, x[r]);
        }
        float nm = fmaxf(m, tm);
        float acc = 0.f;
        #pragma unroll
        for (int r = 0; r < 8; ++r) acc += __expf(x[r] - nm);
        s = s * __expf(m - nm) + acc;
        m = nm;
    }

    // Merge the two half-wave partial softmaxes (rows 0-7 vs 8-15 of each tile).
    float om = __shfl_xor(m, 16, 32);
    float os = __shfl_xor(s, 16, 32);
    float nm = fmaxf(m, om);
    float S  = s * __expf(m - nm) + os * __expf(om - nm);
    float lse = nm + __logf(S);

    if (half == 0) {
        float v = vis[(size_t)n * KP_ + jcol] ? 1.f : 0.f;
        float w = wt[(size_t)n * KP_ + jcol];
        // ce(d1)+ce(d2) are numerically identical forward -> factor 2
        LOSS[(size_t)n * KP_ + jcol] = 2.f * (dj * INV_T_ + lse) * v * w;
    }
}

// ---------------------------------------------------------------------------
// Phase C: deterministic single-block reduction: sum(LOSS) / sum(vis)
// ---------------------------------------------------------------------------
__global__ __launch_bounds__(256) void finalize_kernel(
    const float* __restrict__ LOSS, const unsigned char* __restrict__ vis,
    float* __restrict__ out) {
    __shared__ float sl[256];
    __shared__ float sv[256];
    float accL = 0.f, accV = 0.f;
    for (int i = threadIdx.x; i < NK_; i += 256) {
        accL += LOSS[i];
        accV += vis[i] ? 1.f : 0.f;
    }
    sl[threadIdx.x] = accL;
    sv[threadIdx.x] = accV;
    __syncthreads();
    for (int off = 128; off > 0; off >>= 1) {
        if ((int)threadIdx.x < off) {
            sl[threadIdx.x] += sl[threadIdx.x + off];
            sv[threadIdx.x] += sv[threadIdx.x + off];
        }
        __syncthreads();
    }
    if (threadIdx.x == 0) out[0] = sl[0] / sv[0];
}

extern "C" void kernel_launch(void* const* d_in, const int* in_sizes, int n_in,
                              void* d_out, int out_size, void* d_ws, size_t ws_size,
                              hipStream_t stream) {
    (void)in_sizes; (void)n_in; (void)out_size; (void)ws_size;
    const float*         srcF = (const float*)d_in[0];
    const float*         trgF = (const float*)d_in[1];
    const float*         srcK = (const float*)d_in[2];
    const float*         trgK = (const float*)d_in[3];
    const unsigned char* vis  = (const unsigned char*)d_in[4];  // jnp bool -> 1 byte/elem
    const float*         wt   = (const float*)d_in[5];
    float* outp = (float*)d_out;

    // Workspace (floats): A2[2NK] | Bm[2NK] | PADNA[2NK] | PADNB[2NK] | LOSS[NK]
    float* w     = (float*)d_ws;
    float* A2    = w;
    float* Bm    = w + 2 * (size_t)NK_;
    float* PADNA = w + 4 * (size_t)NK_;
    float* PADNB = w + 6 * (size_t)NK_;
    float* LOSS  = w + 8 * (size_t)NK_;

    sample_kernel<<<(NK_ + 255) / 256, 256, 0, stream>>>(
        srcF, trgF, srcK, trgK, A2, Bm, PADNA, PADNB);

    // 2048 waves (32 batches x 64 column strips), 8 waves/block -> 256 blocks
    pair_lse_kernel<<<256, 256, 0, stream>>>(A2, Bm, PADNA, PADNB, vis, wt, LOSS);

    finalize_kernel<<<1, 256, 0, stream>>>(LOSS, vis, outp);
}